// GraphSageLayerV1_28913719837489
// MI455X (gfx1250) — compile-verified
//
#include <hip/hip_runtime.h>

typedef __attribute__((ext_vector_type(2))) float v2f;
typedef __attribute__((ext_vector_type(8))) float v8f;

namespace {
constexpr int KNBR   = 16;
constexpr int DIN    = 64;
constexpr int DOUT   = 64;
constexpr int DZ     = 128;          // 2*DIN
constexpr int DZP    = 132;          // padded z row: stride mod 64 == 4 -> conflict-free b64 reads
constexpr int TILE_M = 16;
constexpr int WAVES  = 8;            // waves per block (wave32)
constexpr int BLOCK  = WAVES * 32;
constexpr int KSTEPS = DZ / 4;       // 32 WMMA k-steps
constexpr int WB_HI  = KSTEPS * DOUT + 16;  // per-hi region (float2), +16 => hi=1 on opposite banks
}

__global__ __launch_bounds__(BLOCK) void
sage_fused_wmma_f32(const float* __restrict__ x,
                    const int*   __restrict__ nbr,
                    const float* __restrict__ W,
                    const float* __restrict__ bias,
                    float* __restrict__ y,
                    int n_tiles)
{
    __shared__ v2f   Wb[2 * WB_HI];               // ~32.3 KB: pre-swizzled B fragments
    __shared__ float bl[DOUT];
    __shared__ float zl[WAVES * TILE_M * DZP];    // ~66 KB: per-wave z tiles

    const int tid  = threadIdx.x;
    const int wave = tid >> 5;
    const int lane = tid & 31;
    const int nlo  = lane & 15;   // M (A rows) / N (B,C cols) within tile
    const int hi   = lane >> 4;   // half-wave select

    // Stage W as ready-made B fragments: Wb[hi][kk][col] = {W[4kk+2hi][col], W[4kk+2hi+1][col]}.
    for (int i = tid; i < 2 * KSTEPS * DOUT; i += BLOCK) {
        const int h   = i >> 11;              // i / (KSTEPS*DOUT)
        const int r   = i & (KSTEPS * DOUT - 1);
        const int kk  = r >> 6;
        const int col = r & (DOUT - 1);
        const int k   = 4 * kk + 2 * h;
        v2f w;
        w.x = W[k * DOUT + col];
        w.y = W[(k + 1) * DOUT + col];
        Wb[h * WB_HI + kk * DOUT + col] = w;
    }
    if (tid < DOUT) bl[tid] = bias[tid];

    int tile = (int)blockIdx.x * WAVES + wave;
    if (tile >= n_tiles) tile = n_tiles - 1;      // tail waves redo last tile: identical writes
    const long base = (long)tile * TILE_M;

    float* zw = zl + wave * (TILE_M * DZP);
    const int c0 = lane * 2;                      // this lane's 2 feature columns

    // Phase 1: gather + mean-pool + concat -> z[16][128] in LDS.
    for (int m = 0; m < TILE_M; ++m) {
        const long node = base + m;               // wave-uniform
        v2f xs = *(const v2f*)(x + node * DIN + c0);
        const int* nb = nbr + node * KNBR;        // wave-uniform -> scalar loads
        float a0 = 0.f, a1 = 0.f;
        #pragma unroll
        for (int j = 0; j < KNBR; ++j) {
            const float* nr = x + (long)nb[j] * DIN;  // L2-resident gather
            v2f v = *(const v2f*)(nr + c0);
            a0 += v.x;
            a1 += v.y;
        }
        float* zr = zw + m * DZP;
        zr[c0]           = xs.x;
        zr[c0 + 1]       = xs.y;
        zr[DIN + c0]     = a0 * (1.0f / KNBR);
        zr[DIN + c0 + 1] = a1 * (1.0f / KNBR);
    }
    __syncthreads();   // Wb/bias staged + z visible wave/block-wide

    // Phase 2: y[16][64] = z[16][128] @ W[128][64] + b  via V_WMMA_F32_16X16X4_F32.
    // C/D tile: VGPR v holds rows {v, v+8}, col = t*16 + nlo -> init with per-column bias.
    v8f acc[4];
    #pragma unroll
    for (int t = 0; t < 4; ++t) {
        const float bv = bl[t * 16 + nlo];
        #pragma unroll
        for (int v = 0; v < 8; ++v) acc[t][v] = bv;
    }

    // A fragment (16x4 f32): lane<16 -> M=lane, K={k0,k0+1}; lane>=16 -> M=lane-16, K={k0+2,k0+3}.
    const float* za  = zw + nlo * DZP + 2 * hi;
    const v2f*   Wbh = Wb + hi * WB_HI + nlo;     // this lane's B-fragment column
    for (int kk = 0; kk < KSTEPS; ++kk) {
        v2f a = *(const v2f*)(za + 4 * kk);       // single b64, conflict-free (DZP pad)
        const v2f* wrow = Wbh + kk * DOUT;
        #pragma unroll
        for (int t = 0; t < 4; ++t) {
            v2f bf = wrow[t * 16];                // single b64, hi halves on disjoint banks
            acc[t] = __builtin_amdgcn_wmma_f32_16x16x4_f32(
                false, a, false, bf, (short)0, acc[t], false, false);
        }
    }

    // D tile store: VGPR v -> row (v + 8*hi), col t*16 + nlo. Half-wave stores are 64B contiguous.
    #pragma unroll
    for (int t = 0; t < 4; ++t) {
        #pragma unroll
        for (int v = 0; v < 8; ++v) {
            const long row = base + v + 8 * hi;
            y[row * DOUT + t * 16 + nlo] = acc[t][v];
        }
    }
}

extern "C" void kernel_launch(void* const* d_in, const int* in_sizes, int n_in,
                              void* d_out, int out_size, void* d_ws, size_t ws_size,
                              hipStream_t stream) {
    const float* x    = (const float*)d_in[0];   // [N, 64] fp32
    const int*   nbr  = (const int*)d_in[1];     // [N, 16] int32
    const float* W    = (const float*)d_in[2];   // [128, 64] fp32
    const float* bias = (const float*)d_in[3];   // [64] fp32
    float*       y    = (float*)d_out;           // [N, 64] fp32

    const int n_nodes = in_sizes[0] / DIN;                 // 50000
    const int n_tiles = (n_nodes + TILE_M - 1) / TILE_M;   // 3125
    const int blocks  = (n_tiles + WAVES - 1) / WAVES;     // 391

    sage_fused_wmma_f32<<<blocks, BLOCK, 0, stream>>>(x, nbr, W, bias, y, n_tiles);
}